// Agent_25718264169162
// MI455X (gfx1250) — compile-verified
//
#include <hip/hip_runtime.h>
#include <math.h>

typedef __attribute__((ext_vector_type(2))) float v2f;
typedef __attribute__((ext_vector_type(8))) float v8f;

// ---------- helpers ----------

// Order-preserving mapping float -> u32 (monotonic for all finite floats).
__device__ __forceinline__ unsigned ordkey(float f) {
    unsigned u = __float_as_uint(f);
    return (u & 0x80000000u) ? ~u : (u | 0x80000000u);
}

// Wave32 butterfly max-reduce of a packed u64 key (all lanes get the max).
__device__ __forceinline__ unsigned long long wave_max_u64(unsigned long long v) {
    for (int m = 16; m > 0; m >>= 1) {
        unsigned lo = __shfl_xor((unsigned)(v & 0xFFFFFFFFull), m, 32);
        unsigned hi = __shfl_xor((unsigned)(v >> 32), m, 32);
        unsigned long long o = ((unsigned long long)hi << 32) | (unsigned long long)lo;
        if (o > v) v = o;
    }
    return v;
}

// ---------- pass 1: global max ----------

__global__ __launch_bounds__(256) void k_blockmax(const float* __restrict__ x,
                                                  int n4, float* __restrict__ pmax) {
    const float4* x4 = (const float4*)x;
    int gid = blockIdx.x * 256 + threadIdx.x;
    int stride = gridDim.x * 256;
    float m = -INFINITY;
    for (int i = gid; i < n4; i += stride) {
        __builtin_prefetch(&x4[i + stride], 0, 0);
        float4 v = x4[i];
        m = fmaxf(m, fmaxf(fmaxf(v.x, v.y), fmaxf(v.z, v.w)));
    }
    __shared__ float s[256];
    s[threadIdx.x] = m;
    __syncthreads();
    for (int k = 128; k > 0; k >>= 1) {
        if (threadIdx.x < k) s[threadIdx.x] = fmaxf(s[threadIdx.x], s[threadIdx.x + k]);
        __syncthreads();
    }
    if (threadIdx.x == 0) pmax[blockIdx.x] = s[0];
}

__global__ __launch_bounds__(256) void k_finalmax(const float* __restrict__ pmax,
                                                  int n, float* __restrict__ wsf) {
    __shared__ float s[256];
    float m = -INFINITY;
    for (int i = threadIdx.x; i < n; i += 256) m = fmaxf(m, pmax[i]);
    s[threadIdx.x] = m;
    __syncthreads();
    for (int k = 128; k > 0; k >>= 1) {
        if (threadIdx.x < k) s[threadIdx.x] = fmaxf(s[threadIdx.x], s[threadIdx.x + k]);
        __syncthreads();
    }
    if (threadIdx.x == 0) wsf[0] = s[0];
}

// ---------- pass 2: sum of exp, accumulated on the matrix pipe ----------
// Each lane feeds 2 partial sums per iteration into A of a 16x16x4 f32 WMMA
// with B = ones: D = A*1 + C, so D accumulates row-sums (replicated over 16
// columns). Summing the whole accumulator over-counts by exactly 16x.

__global__ __launch_bounds__(256) void k_blocksum(const float* __restrict__ x, int n4,
                                                  const float* __restrict__ wsf,
                                                  float* __restrict__ psum) {
    const float4* x4 = (const float4*)x;
    const float gmax = wsf[0];
    int gid = blockIdx.x * 256 + threadIdx.x;
    int stride = gridDim.x * 256;

    v8f acc = {};
    v2f ones; ones[0] = 1.0f; ones[1] = 1.0f;

    // Uniform trip count -> EXEC is all-1s at every WMMA issue (ISA requirement).
    int iters = n4 / stride;
    for (int it = 0; it < iters; ++it) {
        int i = gid + it * stride;
        __builtin_prefetch(&x4[i + stride], 0, 0);
        float4 v = x4[i];
        v2f a;
        a[0] = __expf(v.x - gmax) + __expf(v.y - gmax);
        a[1] = __expf(v.z - gmax) + __expf(v.w - gmax);
        acc = __builtin_amdgcn_wmma_f32_16x16x4_f32(false, a, false, ones,
                                                    (short)0, acc, false, false);
    }
    float partial = 0.0f;
    for (int j = 0; j < 8; ++j) partial += acc[j];
    partial *= (1.0f / 16.0f);   // undo 16-column replication
    // ragged tail on the VALU (EXEC may be partial here: no WMMA)
    for (int i = gid + iters * stride; i < n4; i += stride) {
        float4 v = x4[i];
        partial += __expf(v.x - gmax) + __expf(v.y - gmax)
                 + __expf(v.z - gmax) + __expf(v.w - gmax);
    }

    __shared__ float s[256];
    s[threadIdx.x] = partial;
    __syncthreads();
    for (int k = 128; k > 0; k >>= 1) {
        if (threadIdx.x < k) s[threadIdx.x] += s[threadIdx.x + k];
        __syncthreads();
    }
    if (threadIdx.x == 0) psum[blockIdx.x] = s[0];
}

__global__ __launch_bounds__(256) void k_finalsum(const float* __restrict__ psum,
                                                  int n, float* __restrict__ wsf) {
    __shared__ float s[256];
    float t = 0.0f;
    for (int i = threadIdx.x; i < n; i += 256) t += psum[i];
    s[threadIdx.x] = t;
    __syncthreads();
    for (int k = 128; k > 0; k >>= 1) {
        if (threadIdx.x < k) s[threadIdx.x] += s[threadIdx.x + k];
        __syncthreads();
    }
    if (threadIdx.x == 0) wsf[1] = 1.0f / s[0];
}

// ---------- prep: column-hit mask + initial actions ----------

__global__ void k_zeromask(unsigned* __restrict__ cm, int n) {
    int i = blockIdx.x * blockDim.x + threadIdx.x;
    if (i < n) cm[i] = 0u;
}

__global__ void k_cont(const int* __restrict__ cont, const int* __restrict__ prev,
                       unsigned* __restrict__ cm, float* __restrict__ outAct, int M) {
    int i = blockIdx.x * blockDim.x + threadIdx.x;
    if (i < M) {
        if (cont[i] != 0) {
            int c = prev[i];
            atomicOr(&cm[c >> 5], 1u << (c & 31));
            outAct[i] = (float)c;
        } else {
            outAct[i] = -1.0f;
        }
    }
}

// ---------- pass 3: write policy + per-row argmax keys ----------

__global__ __launch_bounds__(256) void k_policy_rowmax(
        const float* __restrict__ scores, const int* __restrict__ cont,
        const float* __restrict__ wsf, const unsigned* __restrict__ cm,
        float* __restrict__ policy, unsigned long long* __restrict__ rowkey, int N) {
    int r = blockIdx.x;
    const float gmax = wsf[0], inv = wsf[1];
    const float* srow = scores + (size_t)r * N;
    float*       prow = policy + (size_t)r * N;
    const bool alive = (cont[r] == 0);

    unsigned long long best = 0ull;
    for (int c = threadIdx.x; c < N; c += 256) {
        float p = __expf(srow[c] - gmax) * inv;
        prow[c] = p;
        bool hit = (cm[c >> 5] >> (c & 31)) & 1u;
        if (alive && !hit) {
            unsigned long long key = ((unsigned long long)ordkey(p) << 32)
                                   | ((unsigned long long)(unsigned)c << 16)
                                   | (unsigned)r;
            if (key > best) best = key;
        }
    }
    __shared__ unsigned long long sk[256];
    sk[threadIdx.x] = best;
    __syncthreads();
    for (int k = 128; k > 0; k >>= 1) {
        if (threadIdx.x < k && sk[threadIdx.x + k] > sk[threadIdx.x])
            sk[threadIdx.x] = sk[threadIdx.x + k];
        __syncthreads();
    }
    if (threadIdx.x == 0) rowkey[r] = sk[0];
}

// ---------- pass 4: sequential greedy assignment, one wave32 ----------

#define MAX_M 1024
#define MAX_MASK 128

__global__ __launch_bounds__(32) void k_assign(
        const int* __restrict__ cont, const float* __restrict__ policy,
        const unsigned long long* __restrict__ rowkeyG,
        const unsigned* __restrict__ cmG,
        float* __restrict__ outAct, int M, int N) {
    __shared__ unsigned long long skey[MAX_M];
    __shared__ unsigned scm[MAX_MASK];
    const int lane = threadIdx.x;
    const int nMask = (N + 31) >> 5;

    for (int r = lane; r < MAX_M; r += 32) skey[r] = (r < M) ? rowkeyG[r] : 0ull;
    for (int c = lane; c < MAX_MASK; c += 32) scm[c] = (c < nMask) ? cmG[c] : 0xFFFFFFFFu;
    __syncthreads();

    const int K = (M + 31) >> 5;   // rows handled per lane

    for (int i = 0; i < M; ++i) {
        if (cont[i] != 0) continue;          // wave-uniform

        // argmax over all alive per-row keys
        unsigned long long best = 0ull;
        for (int k = 0; k < K; ++k) {
            int r = lane + 32 * k;
            unsigned long long v = (r < M) ? skey[r] : 0ull;
            if (v > best) best = v;
        }
        best = wave_max_u64(best);
        if (best == 0ull) continue;          // nothing assignable left

        int brow = (int)(best & 0xFFFFull);
        int bcol = (int)((best >> 16) & 0xFFFFull);
        if (lane == 0) {
            outAct[brow] = (float)bcol;
            skey[brow] = 0ull;               // retire row
            scm[bcol >> 5] |= (1u << (bcol & 31));   // retire column
        }
        __syncthreads();

        // rescan rows whose cached best column was just stolen
        for (int k = 0; k < K; ++k) {
            int r = lane + 32 * k;
            unsigned long long v = (r < M) ? skey[r] : 0ull;
            bool aff = (v != 0ull) && ((int)((v >> 16) & 0xFFFFull) == bcol);
            unsigned bal = (unsigned)__ballot(aff);
            while (bal) {
                int l = __ffs(bal) - 1;
                bal &= bal - 1;
                int rr = l + 32 * k;
                const float* prow = policy + (size_t)rr * N;
                unsigned long long nb = 0ull;
                for (int c = lane; c < N; c += 32) {
                    if (scm[c >> 5] & (1u << (c & 31))) continue;
                    float p = prow[c];
                    unsigned long long key = ((unsigned long long)ordkey(p) << 32)
                                           | ((unsigned long long)(unsigned)c << 16)
                                           | (unsigned)rr;
                    if (key > nb) nb = key;
                }
                nb = wave_max_u64(nb);
                if (lane == 0) skey[rr] = nb;
                __syncthreads();
            }
        }
        __syncthreads();
    }
}

// ---------- host launcher ----------

extern "C" void kernel_launch(void* const* d_in, const int* in_sizes, int n_in,
                              void* d_out, int out_size, void* d_ws, size_t ws_size,
                              hipStream_t stream) {
    const float* scores = (const float*)d_in[0];
    const int*   cont   = (const int*)d_in[1];
    const int*   prev   = (const int*)d_in[2];

    const int M = in_sizes[1];
    const int total = in_sizes[0];
    const int N = total / M;
    const int n4 = total / 4;
    const int maskWords = (N + 31) >> 5;

    float* outAct = (float*)d_out;          // M actions (as float)
    float* policy = (float*)d_out + M;      // M*N softmax probabilities

    // workspace layout (floats):
    // [0]=gmax [1]=invsum [8..264)=pmax [264..520)=psum
    // then maskWords u32 colmask, then (8B-aligned) M u64 row keys
    float* wsf  = (float*)d_ws;
    float* pmax = wsf + 8;
    float* psum = wsf + 8 + 256;
    unsigned* colmask = (unsigned*)(wsf + 8 + 512);
    size_t off = 8 + 512 + (size_t)maskWords;
    off = (off + 1) & ~(size_t)1;           // 8-byte align for u64
    unsigned long long* rowkey = (unsigned long long*)(wsf + off);

    const int RB = 256;                      // reduction grid

    k_blockmax<<<RB, 256, 0, stream>>>(scores, n4, pmax);
    k_finalmax<<<1, 256, 0, stream>>>(pmax, RB, wsf);
    k_blocksum<<<RB, 256, 0, stream>>>(scores, n4, wsf, psum);
    k_finalsum<<<1, 256, 0, stream>>>(psum, RB, wsf);

    k_zeromask<<<(maskWords + 255) / 256, 256, 0, stream>>>(colmask, maskWords);
    k_cont<<<(M + 255) / 256, 256, 0, stream>>>(cont, prev, colmask, outAct, M);

    k_policy_rowmax<<<M, 256, 0, stream>>>(scores, cont, wsf, colmask, policy, rowkey, N);
    k_assign<<<1, 32, 0, stream>>>(cont, policy, rowkey, colmask, outAct, M, N);
}